// TemporalGNN_13898514170736
// MI455X (gfx1250) — compile-verified
//
#include <hip/hip_runtime.h>

#define KN 100000
#define KE 1600000
#define KM (KE / 2)

typedef __attribute__((ext_vector_type(2))) float v2f;
typedef __attribute__((ext_vector_type(8))) float v8f;
typedef __attribute__((ext_vector_type(4))) unsigned u32x4;
typedef __attribute__((ext_vector_type(8))) unsigned u32x8;

// ---------- helpers ----------
__device__ __forceinline__ unsigned ordf(float f) {
  unsigned u = __float_as_uint(f);
  return (u & 0x80000000u) ? ~u : (u | 0x80000000u);  // order-preserving map
}
__device__ __forceinline__ float unordf(unsigned t) {
  unsigned u = (t & 0x80000000u) ? (t ^ 0x80000000u) : ~t;
  return __uint_as_float(u);
}
__device__ __forceinline__ float wave_sum(float p) {
#pragma unroll
  for (int m = 16; m > 0; m >>= 1) p += __shfl_xor(p, m, 32);
  return p;
}

// ---------- Tensor Data Mover: 2D tile global -> LDS ----------
// D# per CDNA5 ISA. Rows beyond rows_avail are OOB and
// zero-filled by the TDM (free tail padding). Issue from one wave, then
// s_wait_tensorcnt + workgroup barrier before LDS consumption.
__device__ __forceinline__ void tdm_load_2d(const void* gsrc, void* lds_dst,
                                            unsigned width_elts,
                                            unsigned rows_tile,
                                            unsigned rows_avail,
                                            unsigned stride_elts) {
  unsigned long long ga = (unsigned long long)gsrc;
  unsigned lds = (unsigned)(unsigned long long)lds_dst;  // addr[31:0] = LDS byte offset
  u32x4 g0;
  g0[0] = 1u;                                            // count=1, user mode
  g0[1] = lds;                                           // lds_addr
  g0[2] = (unsigned)ga;                                  // global_addr[31:0]
  g0[3] = (unsigned)((ga >> 32) & 0x01FFFFFFull) | 0x80000000u;  // addr[56:32] | type=2
  u32x8 g1;
  g1[0] = 0x20000u;                                      // data_size=2 (4 bytes)
  g1[1] = (width_elts & 0xFFFFu) << 16;                  // tensor_dim0[15:0]
  g1[2] = (width_elts >> 16) | ((rows_avail & 0xFFFFu) << 16);  // dim0 hi | dim1 lo
  g1[3] = (rows_avail >> 16) | ((width_elts & 0xFFFFu) << 16);  // dim1 hi | tile_dim0
  g1[4] = rows_tile & 0xFFFFu;                           // tile_dim1 (tile_dim2=0)
  g1[5] = stride_elts;                                   // tensor_dim0_stride[31:0]
  g1[6] = 0u;
  g1[7] = 0u;
  asm volatile("tensor_load_to_lds %0, %1" ::"s"(g0), "s"(g1) : "memory");
}

// ---------- fused 2-layer MLP: out = relu(in@W1+b1)@W2+b2 ----------
// 128 threads (4 waves), 64 rows/block, wave owns 16x64 tile via f32 WMMA.
// Input tile (and weights when STAGEW) staged by the Tensor Data Mover.
template <int K1, bool STAGEW>
__global__ __launch_bounds__(128) void mlp2_kernel(
    const float* __restrict__ in, const float* __restrict__ W1,
    const float* __restrict__ b1, const float* __restrict__ W2,
    const float* __restrict__ b2, float* __restrict__ out, int R) {
  __shared__ __align__(16) float sIn[64 * K1];
  __shared__ __align__(16) float sHid[64 * 64];
  __shared__ __align__(16) float sW1[STAGEW ? K1 * 64 : 1];
  __shared__ __align__(16) float sW2[STAGEW ? 64 * 64 : 1];
  const int tid = threadIdx.x;
  const long long row0 = (long long)blockIdx.x * 64;

  if (tid < 32) {                  // wave0: input tile (OOB rows -> 0)
    tdm_load_2d(in + row0 * K1, sIn, K1, 64, (unsigned)(R - row0), K1);
    __builtin_amdgcn_s_wait_tensorcnt(0);
  } else if (STAGEW && tid < 64) { // wave1: W1
    tdm_load_2d(W1, sW1, 64, K1, K1, 64);
    __builtin_amdgcn_s_wait_tensorcnt(0);
  } else if (STAGEW && tid < 96) { // wave2: W2
    tdm_load_2d(W2, sW2, 64, 64, 64, 64);
    __builtin_amdgcn_s_wait_tensorcnt(0);
  }
  __syncthreads();

  const int lane = tid & 31;
  const int r16 = (tid >> 5) << 4;
  const int lr = lane & 15;
  const int kb = (lane >> 4) << 1;
  const int rb = r16 + ((lane >> 4) << 3);
  for (int ct = 0; ct < 4; ++ct) {
    v8f acc = {};
    const int col = ct * 16 + lr;
    for (int k = 0; k < K1; k += 4) {
      v2f a, b;
      a.x = sIn[(r16 + lr) * K1 + k + kb];
      a.y = sIn[(r16 + lr) * K1 + k + kb + 1];
      if constexpr (STAGEW) {
        b.x = sW1[(k + kb) * 64 + col];
        b.y = sW1[(k + kb + 1) * 64 + col];
      } else {
        b.x = W1[(k + kb) * 64 + col];
        b.y = W1[(k + kb + 1) * 64 + col];
      }
      acc = __builtin_amdgcn_wmma_f32_16x16x4_f32(false, a, false, b, (short)0,
                                                  acc, false, false);
    }
    const float bb = b1[col];
#pragma unroll
    for (int i = 0; i < 8; ++i) {
      float v = acc[i] + bb;
      sHid[(rb + i) * 64 + col] = v > 0.0f ? v : 0.0f;
    }
  }
  __syncthreads();
  for (int ct = 0; ct < 4; ++ct) {
    v8f acc = {};
    const int col = ct * 16 + lr;
    for (int k = 0; k < 64; k += 4) {
      v2f a, b;
      a.x = sHid[(r16 + lr) * 64 + k + kb];
      a.y = sHid[(r16 + lr) * 64 + k + kb + 1];
      if constexpr (STAGEW) {
        b.x = sW2[(k + kb) * 64 + col];
        b.y = sW2[(k + kb + 1) * 64 + col];
      } else {
        b.x = W2[(k + kb) * 64 + col];
        b.y = W2[(k + kb + 1) * 64 + col];
      }
      acc = __builtin_amdgcn_wmma_f32_16x16x4_f32(false, a, false, b, (short)0,
                                                  acc, false, false);
    }
    const float bb = b2[col];
#pragma unroll
    for (int i = 0; i < 8; ++i) {
      long long gr = row0 + rb + i;
      if (gr < R) out[gr * 64 + col] = acc[i] + bb;
    }
  }
}

// ---------- xw = h@W, plus a_src = xw@a_s, a_dst = xw@a_d ----------
__global__ __launch_bounds__(128) void gat_xw_kernel(
    const float* __restrict__ h, const float* __restrict__ W,
    const float* __restrict__ a_s, const float* __restrict__ a_d,
    float* __restrict__ xw, float* __restrict__ a_src,
    float* __restrict__ a_dst) {
  __shared__ __align__(16) float sH[64 * 64];
  __shared__ __align__(16) float sX[64 * 64];
  __shared__ __align__(16) float sW[64 * 64];
  const int tid = threadIdx.x;
  const long long row0 = (long long)blockIdx.x * 64;

  if (tid < 32) {                  // wave0: h tile (OOB rows -> 0)
    tdm_load_2d(h + row0 * 64, sH, 64, 64, (unsigned)(KN - row0), 64);
    __builtin_amdgcn_s_wait_tensorcnt(0);
  } else if (tid < 64) {           // wave1: W
    tdm_load_2d(W, sW, 64, 64, 64, 64);
    __builtin_amdgcn_s_wait_tensorcnt(0);
  }
  __syncthreads();

  const int lane = tid & 31;
  const int r16 = (tid >> 5) << 4;
  const int lr = lane & 15;
  const int kb = (lane >> 4) << 1;
  const int rb = r16 + ((lane >> 4) << 3);
  for (int ct = 0; ct < 4; ++ct) {
    v8f acc = {};
    const int col = ct * 16 + lr;
    for (int k = 0; k < 64; k += 4) {
      v2f a, b;
      a.x = sH[(r16 + lr) * 64 + k + kb];
      a.y = sH[(r16 + lr) * 64 + k + kb + 1];
      b.x = sW[(k + kb) * 64 + col];
      b.y = sW[(k + kb + 1) * 64 + col];
      acc = __builtin_amdgcn_wmma_f32_16x16x4_f32(false, a, false, b, (short)0,
                                                  acc, false, false);
    }
#pragma unroll
    for (int i = 0; i < 8; ++i) {
      sX[(rb + i) * 64 + col] = acc[i];
      long long gr = row0 + rb + i;
      if (gr < KN) xw[gr * 64 + col] = acc[i];
    }
  }
  __syncthreads();
  if (tid < 64) {
    long long gr = row0 + tid;
    if (gr < KN) {
      float ps = 0.0f, pd = 0.0f;
      for (int c = 0; c < 64; ++c) {
        float v = sX[tid * 64 + c];
        ps += v * a_s[c];
        pd += v * a_d[c];
      }
      a_src[gr] = ps;
      a_dst[gr] = pd;
    }
  }
}

// ---------- wv = We @ a_e (64x64 * 64) ----------
__global__ void wv_kernel(const float* __restrict__ We,
                          const float* __restrict__ a_e,
                          float* __restrict__ wv) {
  int i = threadIdx.x;
  float s = 0.0f;
  for (int j = 0; j < 64; ++j) s += We[i * 64 + j] * a_e[j];
  wv[i] = s;
}

// ---------- in-degree count (layer-invariant) ----------
__global__ __launch_bounds__(256) void count_kernel(const int* __restrict__ dst,
                                                    float* __restrict__ cnt) {
  long long e = (long long)blockIdx.x * 256 + threadIdx.x;
  if (e < KE) atomicAdd(&cnt[dst[e]], 1.0f);
}

// ---------- pass A: a_edge[e]=dot(ea[e],wv); sum_ae[dst]+=a_edge ----------
__global__ __launch_bounds__(256) void edge_aedge_kernel(
    const float* __restrict__ ea64, const int* __restrict__ dst,
    const float* __restrict__ wv, float* __restrict__ aebuf,
    float* __restrict__ sum_ae) {
  const long long wid = (long long)blockIdx.x * 8 + (threadIdx.x >> 5);
  const int lane = threadIdx.x & 31;
  if (wid >= KE) return;
  const float* row = ea64 + wid * 64;
  float p = row[lane] * wv[lane] + row[lane + 32] * wv[lane + 32];
  p = wave_sum(p);
  if (lane == 0) {
    aebuf[wid] = p;
    atomicAdd(&sum_ae[dst[wid]], p);
  }
}

// ---------- pass B1: self-loop alpha, init segment max ----------
__global__ __launch_bounds__(256) void node_loop_alpha_kernel(
    const float* __restrict__ a_src, const float* __restrict__ a_dst,
    const float* __restrict__ sum_ae, const float* __restrict__ cnt,
    float* __restrict__ alpha_loop, unsigned* __restrict__ m_u) {
  int i = blockIdx.x * 256 + threadIdx.x;
  if (i >= KN) return;
  float c = cnt[i];
  c = c > 1.0f ? c : 1.0f;
  float a = a_src[i] + a_dst[i] + sum_ae[i] / c;  // linear: dot(mean ea, wv)
  a = a > 0.0f ? a : 0.2f * a;                    // leaky_relu 0.2
  alpha_loop[i] = a;
  m_u[i] = ordf(a);
}

// ---------- pass B2: edge alpha + segment max ----------
__global__ __launch_bounds__(256) void edge_alpha_kernel(
    const int* __restrict__ src, const int* __restrict__ dst,
    const float* __restrict__ a_src, const float* __restrict__ a_dst,
    float* __restrict__ aebuf, unsigned* __restrict__ m_u) {
  long long e = (long long)blockIdx.x * 256 + threadIdx.x;
  if (e >= KE) return;
  int s = src[e], d = dst[e];
  float a = a_src[s] + a_dst[d] + aebuf[e];
  a = a > 0.0f ? a : 0.2f * a;
  aebuf[e] = a;  // overwrite a_edge with alpha
  atomicMax(&m_u[d], ordf(a));
}

// ---------- pass C: numer[dst]+=ex*xw[src], den[dst]+=ex ----------
__global__ __launch_bounds__(256) void edge_accum_kernel(
    const int* __restrict__ src, const int* __restrict__ dst,
    const float* __restrict__ aebuf, const unsigned* __restrict__ m_u,
    const float* __restrict__ xw, float* __restrict__ numer,
    float* __restrict__ den) {
  const long long wid = (long long)blockIdx.x * 8 + (threadIdx.x >> 5);
  const int lane = threadIdx.x & 31;
  if (wid >= KE) return;
  int s = src[wid], d = dst[wid];
  float ex = __expf(aebuf[wid] - unordf(m_u[d]));
  const float* xr = xw + (long long)s * 64;
  float* nr = numer + (long long)d * 64;
  atomicAdd(&nr[lane], ex * xr[lane]);
  atomicAdd(&nr[lane + 32], ex * xr[lane + 32]);
  if (lane == 0) atomicAdd(&den[d], ex);
}

// ---------- pass D: add self-loop term, normalize, bias, relu ----------
__global__ __launch_bounds__(256) void node_finalize_kernel(
    const float* __restrict__ alpha_loop, const unsigned* __restrict__ m_u,
    const float* __restrict__ den, const float* __restrict__ numer,
    const float* __restrict__ xw, const float* __restrict__ bias,
    float* __restrict__ hout, int do_relu) {
  const long long i = (long long)blockIdx.x * 8 + (threadIdx.x >> 5);
  const int lane = threadIdx.x & 31;
  if (i >= KN) return;
  float exl = __expf(alpha_loop[i] - unordf(m_u[i]));
  float dn = den[i] + exl;
  const float* nr = numer + i * 64;
  const float* xr = xw + i * 64;
#pragma unroll
  for (int half = 0; half < 2; ++half) {
    int c = lane + half * 32;
    float v = (nr[c] + exl * xr[c]) / dn + bias[c];
    if (do_relu) v = v > 0.0f ? v : 0.0f;
    hout[i * 64 + c] = v;
  }
}

// ---------- edge predictor: pred = relu(cat(h[se],h[de])@W1+b1)@w2+b2 ----------
__global__ __launch_bounds__(128) void edge_pred_kernel(
    const float* __restrict__ h, const int* __restrict__ src,
    const int* __restrict__ dst, const float* __restrict__ W1,
    const float* __restrict__ b1, const float* __restrict__ w2,
    const float* __restrict__ b2, float* __restrict__ out) {
  __shared__ __align__(16) float sF[64 * 128];
  __shared__ __align__(16) float sHid[64 * 64];
  __shared__ float sw2[64];
  const int tid = threadIdx.x;
  const long long row0 = (long long)blockIdx.x * 64;
  if (tid < 64) sw2[tid] = w2[tid];
  for (int i = tid; i < 64 * 128; i += 128) {
    int r = i >> 7, c = i & 127;
    long long gr = row0 + r;
    float v = 0.0f;
    if (gr < KM) {
      long long e = gr * 2;  // even edges only
      int node = (c < 64) ? src[e] : dst[e];
      v = h[(long long)node * 64 + (c & 63)];
    }
    sF[i] = v;
  }
  __syncthreads();
  const int lane = tid & 31;
  const int r16 = (tid >> 5) << 4;
  const int lr = lane & 15;
  const int kb = (lane >> 4) << 1;
  const int rb = r16 + ((lane >> 4) << 3);
  for (int ct = 0; ct < 4; ++ct) {
    v8f acc = {};
    const int col = ct * 16 + lr;
    for (int k = 0; k < 128; k += 4) {
      v2f a, b;
      a.x = sF[(r16 + lr) * 128 + k + kb];
      a.y = sF[(r16 + lr) * 128 + k + kb + 1];
      b.x = W1[(k + kb) * 64 + col];
      b.y = W1[(k + kb + 1) * 64 + col];
      acc = __builtin_amdgcn_wmma_f32_16x16x4_f32(false, a, false, b, (short)0,
                                                  acc, false, false);
    }
    const float bb = b1[col];
#pragma unroll
    for (int i = 0; i < 8; ++i) {
      float v = acc[i] + bb;
      sHid[(rb + i) * 64 + col] = v > 0.0f ? v : 0.0f;
    }
  }
  __syncthreads();
  if (tid < 64) {
    long long gr = row0 + tid;
    if (gr < KM) {
      float s = 0.0f;
      for (int c = 0; c < 64; ++c) s += sHid[tid * 64 + c] * sw2[c];
      s += b2[0];
      out[gr * 2] = s;
      out[gr * 2 + 1] = s;
    }
  }
}

// ---------- host orchestration ----------
extern "C" void kernel_launch(void* const* d_in, const int* in_sizes, int n_in,
                              void* d_out, int out_size, void* d_ws,
                              size_t ws_size, hipStream_t stream) {
  (void)in_sizes; (void)n_in; (void)out_size; (void)ws_size;
  const float* x = (const float*)d_in[0];
  const float* edge_attr = (const float*)d_in[1];
  const int* edge_index = (const int*)d_in[2];
  const float* ne_w1 = (const float*)d_in[3];
  const float* ne_b1 = (const float*)d_in[4];
  const float* ne_w2 = (const float*)d_in[5];
  const float* ne_b2 = (const float*)d_in[6];
  const float* ee_w1 = (const float*)d_in[7];
  const float* ee_b1 = (const float*)d_in[8];
  const float* ee_w2 = (const float*)d_in[9];
  const float* ee_b2 = (const float*)d_in[10];
  const float* ep_w1 = (const float*)d_in[11];
  const float* ep_b1 = (const float*)d_in[12];
  const float* ep_w2 = (const float*)d_in[13];
  const float* ep_b2 = (const float*)d_in[14];
  const int* src = edge_index;
  const int* dst = edge_index + KE;

  float* ws = (float*)d_ws;
  size_t off = 0;
  auto alloc = [&](size_t nf) {
    float* p = ws + off;
    off += (nf + 63) & ~(size_t)63;
    return p;
  };
  float* h0 = alloc((size_t)KN * 64);
  float* h1 = alloc((size_t)KN * 64);
  float* ea64 = alloc((size_t)KE * 64);
  float* xw = alloc((size_t)KN * 64);
  float* numer = alloc((size_t)KN * 64);
  float* a_srcb = alloc(KN);
  float* a_dstb = alloc(KN);
  float* cntb = alloc(KN);
  float* sum_ae = alloc(KN);
  float* alpha_l = alloc(KN);
  float* denb = alloc(KN);
  unsigned* m_u = (unsigned*)alloc(KN);
  float* aebuf = alloc(KE);
  float* wvb = alloc(64);

  const int nodeBlocks = (KN + 63) / 64;
  const int edgeRowBlocks = (KE + 63) / 64;
  const int predBlocks = (KM + 63) / 64;
  const int edgeWaveBlocks = (KE + 7) / 8;
  const int nodeWaveBlocks = (KN + 7) / 8;
  const int edgeThrBlocks = (KE + 255) / 256;
  const int nodeThrBlocks = (KN + 255) / 256;

  // encoders
  mlp2_kernel<128, false><<<nodeBlocks, 128, 0, stream>>>(x, ne_w1, ne_b1,
                                                          ne_w2, ne_b2, h0, KN);
  mlp2_kernel<32, true><<<edgeRowBlocks, 128, 0, stream>>>(
      edge_attr, ee_w1, ee_b1, ee_w2, ee_b2, ea64, KE);
  hipMemsetAsync(cntb, 0, KN * sizeof(float), stream);
  count_kernel<<<edgeThrBlocks, 256, 0, stream>>>(dst, cntb);

  const float* hin = h0;
  float* hbufs[2] = {h1, h0};
  for (int l = 0; l < 3; ++l) {
    const float* gw = (const float*)d_in[15 + 6 * l + 0];
    const float* gas = (const float*)d_in[15 + 6 * l + 1];
    const float* gad = (const float*)d_in[15 + 6 * l + 2];
    const float* gwe = (const float*)d_in[15 + 6 * l + 3];
    const float* gae = (const float*)d_in[15 + 6 * l + 4];
    const float* gb = (const float*)d_in[15 + 6 * l + 5];
    float* hout = hbufs[l & 1];
    hipMemsetAsync(sum_ae, 0, KN * sizeof(float), stream);
    hipMemsetAsync(denb, 0, KN * sizeof(float), stream);
    hipMemsetAsync(numer, 0, (size_t)KN * 64 * sizeof(float), stream);
    wv_kernel<<<1, 64, 0, stream>>>(gwe, gae, wvb);
    gat_xw_kernel<<<nodeBlocks, 128, 0, stream>>>(hin, gw, gas, gad, xw,
                                                  a_srcb, a_dstb);
    edge_aedge_kernel<<<edgeWaveBlocks, 256, 0, stream>>>(ea64, dst, wvb,
                                                          aebuf, sum_ae);
    node_loop_alpha_kernel<<<nodeThrBlocks, 256, 0, stream>>>(
        a_srcb, a_dstb, sum_ae, cntb, alpha_l, m_u);
    edge_alpha_kernel<<<edgeThrBlocks, 256, 0, stream>>>(src, dst, a_srcb,
                                                         a_dstb, aebuf, m_u);
    edge_accum_kernel<<<edgeWaveBlocks, 256, 0, stream>>>(src, dst, aebuf, m_u,
                                                          xw, numer, denb);
    node_finalize_kernel<<<nodeWaveBlocks, 256, 0, stream>>>(
        alpha_l, m_u, denb, numer, xw, gb, hout, l < 2 ? 1 : 0);
    hin = hout;
  }
  edge_pred_kernel<<<predBlocks, 128, 0, stream>>>(hin, src, dst, ep_w1, ep_b1,
                                                   ep_w2, ep_b2, (float*)d_out);
}